// BoxHungarianMatcher_21655225106651
// MI455X (gfx1250) — compile-verified
//
#include <hip/hip_runtime.h>

typedef __attribute__((ext_vector_type(2))) float v2f;
typedef __attribute__((ext_vector_type(8))) float v8f;

#define NQTOT 14400   // BS*NQ
#define NTGT  1600
#define NCLS  80

#define W_CLASS 2.0f
#define W_BBOX  5.0f
#define W_GIOU  2.0f

// workspace layout (float offsets)
#define TABLE_OFF 0                       // 14400*80       = 1,152,000 f
#define PREDF_OFF (NQTOT * NCLS)          // 14400*12       =   172,800 f
#define TGTF_OFF  (PREDF_OFF + NQTOT*12)  // 1600*12        =    19,200 f
// total = 1,344,000 floats = 5.376 MB

#define JCHUNK 400
#define WAVES  4

// ---------------------------------------------------------------------------
// Prep 1: focal class-cost table, one thread per (query, class) entry.
// cc(i,c) = ALPHA*(1-p)^2*(-log(p+eps)) - (1-ALPHA)*p^2*(-log(1-p+eps))
// ---------------------------------------------------------------------------
__global__ void prep_class_kernel(const float* __restrict__ logits,
                                  float* __restrict__ table) {
    int idx = blockIdx.x * 256 + threadIdx.x;
    if (idx >= NQTOT * NCLS) return;
    float x = logits[idx];
    float p = 1.0f / (1.0f + __expf(-x));
    float pos = 0.25f * (1.0f - p) * (1.0f - p) * (-__logf(p + 1e-8f));
    float neg = 0.75f * p * p * (-__logf(1.0f - p + 1e-8f));
    table[idx] = pos - neg;
}

// ---------------------------------------------------------------------------
// Prep 2: box features (xyxy, area, cxcywh) for preds and targets.
// Row stride 12 floats (48 B) so rows are float4-aligned.
// ---------------------------------------------------------------------------
__device__ __forceinline__ void boxFeat(const float* __restrict__ b,
                                        float* __restrict__ r) {
    float cx = b[0], cy = b[1], w = b[2], h = b[3];
    float x0 = cx - 0.5f * w, y0 = cy - 0.5f * h;
    float x1 = cx + 0.5f * w, y1 = cy + 0.5f * h;
    r[0] = x0; r[1] = y0; r[2] = x1; r[3] = y1;
    r[4] = (x1 - x0) * (y1 - y0);   // area from xyxy, like the reference
    r[5] = cx; r[6] = cy; r[7] = w; r[8] = h;
    r[9] = 0.f; r[10] = 0.f; r[11] = 0.f;
}

__global__ void prep_boxes_kernel(const float* __restrict__ pb,
                                  const float* __restrict__ tb,
                                  float* __restrict__ predF,
                                  float* __restrict__ tgtF) {
    int i = blockIdx.x * 256 + threadIdx.x;
    if (i < NQTOT) boxFeat(pb + i * 4, predF + i * 12);
    if (i < NTGT)  boxFeat(tb + i * 4, tgtF + i * 12);
}

// ---------------------------------------------------------------------------
// Main: 4 waves/block, one 16-query tile per wave, 400 targets per block
// (grid.y selects the j-chunk). Class cost via one-hot V_WMMA_F32_16X16X4_F32;
// L1 + GIoU on VALU; combined and stored per D-matrix layout.
// ---------------------------------------------------------------------------
__global__ void __launch_bounds__(128)
matcher_cost_kernel(const float* __restrict__ table,
                    const float* __restrict__ predF,
                    const float* __restrict__ tgtF,
                    const int*   __restrict__ labels,
                    float*       __restrict__ out) {
    __shared__ float sT[JCHUNK * 12];
    __shared__ int   sL[JCHUNK];

    const int jbase = blockIdx.y * JCHUNK;
    for (int t = threadIdx.x; t < JCHUNK * 12; t += 128) sT[t] = tgtF[jbase * 12 + t];
    for (int t = threadIdx.x; t < JCHUNK; t += 128)      sL[t] = labels[jbase + t];
    __syncthreads();

    const int wave    = threadIdx.x >> 5;
    const int lane    = threadIdx.x & 31;
    const int laneLo  = lane & 15;
    const int hiHalf  = lane >> 4;       // 0: lanes 0-15, 1: lanes 16-31
    const int base2   = hiHalf << 1;     // A/B layout: hi half holds K+2
    const int base8   = hiHalf << 3;     // D layout:   hi half holds M+8
    const int itile   = (blockIdx.x * WAVES + wave) << 4;

    // A matrix (16 queries x 80 classes of class-cost), ISA 16x4 f32 layout:
    // A[c] VGPR0 = K=4c+base2, VGPR1 = K=4c+base2+1, M = laneLo.
    v2f A[20];
    {
        const float* ct = table + (itile + laneLo) * NCLS + base2;
#pragma unroll
        for (int c = 0; c < 20; ++c) { A[c].x = ct[4 * c]; A[c].y = ct[4 * c + 1]; }
    }

    // Pred features for this lane's 8 output rows (M = base8 + r).
    float4 pXY[8], pAC[8];
    float  pH[8];
#pragma unroll
    for (int r = 0; r < 8; ++r) {
        const float4* pr = (const float4*)(predF + (size_t)(itile + base8 + r) * 12);
        pXY[r] = pr[0];                       // x0 y0 x1 y1
        pAC[r] = pr[1];                       // area cx cy w
        pH[r]  = ((const float*)pr)[8];       // h
    }

    for (int jt = 0; jt < JCHUNK / 16; ++jt) {
        const int j0 = jt * 16;

        const float* tf = &sT[(j0 + laneLo) * 12];
        float tx0 = tf[0], ty0 = tf[1], tx1 = tf[2], ty1 = tf[3];
        float tarea = tf[4], tcx = tf[5], tcy = tf[6], tw = tf[7], th = tf[8];
        int   lab = sL[j0 + laneLo];

        // One-hot class-select GEMM tile: 20 K-steps, two accumulator chains.
        v8f acc0 = {}, acc1 = {};
#pragma unroll
        for (int c = 0; c < 20; c += 2) {
            int k0 = 4 * c + base2;
            v2f B0; B0.x = (lab == k0)     ? 1.0f : 0.0f;
                    B0.y = (lab == k0 + 1) ? 1.0f : 0.0f;
            acc0 = __builtin_amdgcn_wmma_f32_16x16x4_f32(
                false, A[c], false, B0, (short)0, acc0, false, false);
            int k1 = 4 * (c + 1) + base2;
            v2f B1; B1.x = (lab == k1)     ? 1.0f : 0.0f;
                    B1.y = (lab == k1 + 1) ? 1.0f : 0.0f;
            acc1 = __builtin_amdgcn_wmma_f32_16x16x4_f32(
                false, A[c + 1], false, B1, (short)0, acc1, false, false);
        }

#pragma unroll
        for (int r = 0; r < 8; ++r) {
            float px0 = pXY[r].x, py0 = pXY[r].y, px1 = pXY[r].z, py1 = pXY[r].w;
            float parea = pAC[r].x, pcx = pAC[r].y, pcy = pAC[r].z, pw = pAC[r].w;
            float ph = pH[r];

            // intersection
            float iw = fminf(px1, tx1) - fmaxf(px0, tx0);
            float ih = fminf(py1, ty1) - fmaxf(py0, ty0);
            iw = fmaxf(iw, 0.0f); ih = fmaxf(ih, 0.0f);
            float inter = iw * ih;
            float uni = parea + tarea - inter;
            float iou = inter * __builtin_amdgcn_rcpf(uni);
            // enclosing box
            float ew = fmaxf(px1, tx1) - fminf(px0, tx0);
            float eh = fmaxf(py1, ty1) - fminf(py0, ty0);
            ew = fmaxf(ew, 0.0f); eh = fmaxf(eh, 0.0f);
            float ae = ew * eh;
            float giou = iou - (ae - uni) * __builtin_amdgcn_rcpf(ae);
            // L1 on cxcywh
            float l1 = fabsf(pcx - tcx) + fabsf(pcy - tcy) +
                       fabsf(pw - tw)   + fabsf(ph - th);
            float ccls = acc0[r] + acc1[r];
            float cost = W_BBOX * l1 + W_CLASS * ccls - W_GIOU * giou;

            out[(size_t)(itile + base8 + r) * NTGT + jbase + j0 + laneLo] = cost;
        }
    }
}

// ---------------------------------------------------------------------------
extern "C" void kernel_launch(void* const* d_in, const int* in_sizes, int n_in,
                              void* d_out, int out_size, void* d_ws, size_t ws_size,
                              hipStream_t stream) {
    const float* logits = (const float*)d_in[0];   // (16,900,80)
    const float* pboxes = (const float*)d_in[1];   // (16,900,4)
    const int*   labels = (const int*)d_in[2];     // (1600,)
    const float* tboxes = (const float*)d_in[3];   // (1600,4)
    float* ws    = (float*)d_ws;
    float* table = ws + TABLE_OFF;
    float* predF = ws + PREDF_OFF;
    float* tgtF  = ws + TGTF_OFF;
    float* out   = (float*)d_out;

    prep_class_kernel<<<(NQTOT * NCLS + 255) / 256, 256, 0, stream>>>(logits, table);
    prep_boxes_kernel<<<(NQTOT + 255) / 256, 256, 0, stream>>>(pboxes, tboxes, predF, tgtF);
    matcher_cost_kernel<<<dim3(NQTOT / 16 / WAVES, NTGT / JCHUNK), 128, 0, stream>>>(
        table, predF, tgtF, labels, out);
}